// Solver_85461259256252
// MI455X (gfx1250) — compile-verified
//
#include <hip/hip_runtime.h>
#include <hip/hip_bf16.h>

// Prefix-application scan: res[k] = x @ (m_0 @ ... @ m_{k-1}), k = 0..N.
// N = 1024 matrices, D = 256, fp32 throughout (precision matters for a
// 1024-long near-identity product chain -> use V_WMMA_F32_16X16X4_F32).
//
// Chunked scan: C = 64 chunks of G = 16 matrices.
//   Phase 1: per-chunk products P_j = m_{16j} @ ... @ m_{16j+15} via WMMA,
//            15 stream-ordered launches (launch ordering = global sync).
//   Phase 2: single-WG serial GEMV scan over the 64 chunk products (L2-warm).
//   Phase 3: 64 parallel WGs expand each chunk with a 15-step GEMV chain.

#define D      256
#define NMAT   1024
#define G      16
#define NCHUNK 64
#define MATELT (D * D)            // 65536 floats per matrix
#define CHUNKSTRIDE (2 * MATELT)  // ping + pong per chunk

typedef __attribute__((ext_vector_type(2))) float v2f;
typedef __attribute__((ext_vector_type(8))) float v8f;

// ---------------------------------------------------------------------------
// Phase 1: one multiply step t (1..15) of every chunk's left-to-right product.
//   dst(t) = (t odd) ? buf0 : buf1 ; src(t) = dst(t-1), src(1) = m_{16j}.
//   grid = (64 chunks, 2 slabs of 128 rows); block = 256 threads = 8 waves.
//   Each wave owns a 4x4 block of 16x16 tiles (64x64 of C) -> 16
//   v_wmma_f32_16x16x4_f32 per K-step against 12 vector loads
//   (4x b64 A-frags + 8x b32 B-frags); 128 accumulator VGPRs, no spill.
// ---------------------------------------------------------------------------
__global__ __launch_bounds__(256) void prodscan_chunk_mm(
    const float* __restrict__ m, float* __restrict__ ws, int t) {
  const int j    = blockIdx.x;   // chunk
  const int slab = blockIdx.y;   // 0..1 -> rows [slab*128, slab*128+128)

  float* buf0 = ws + (size_t)j * CHUNKSTRIDE;
  float* buf1 = buf0 + MATELT;

  const float* A = (t == 1) ? (m + (size_t)j * G * MATELT)
                            : ((t & 1) ? buf1 : buf0);
  const float* B = m + (size_t)(j * G + t) * MATELT;
  float*       C = (t & 1) ? buf0 : buf1;

  const int w     = threadIdx.x >> 5;   // wave 0..7
  const int lane  = threadIdx.x & 31;
  const int laneM = lane & 15;          // row/col within 16
  const int laneH = lane >> 4;          // half-select (K or M offset)

  // Wave's 4x4 tile block inside the 8x16 tile slab.
  const int tr0 = slab * 8 + (w >> 2) * 4;  // global tile-row base (4 rows)
  const int tc0 = (w & 3) * 4;              // tile-col base (4 cols)

  v8f acc[4][4] = {};  // 128 accumulator VGPRs

  for (int k0 = 0; k0 < D; k0 += 4) {
    v2f af[4];
    v2f bf[4];
    // A 16x4 fragment (ISA 7.12.2): lane M = laneM; VGPR0 = K {0,2}[laneH],
    // VGPR1 = K {1,3}[laneH] -> contiguous pair -> one b64 load.
#pragma unroll
    for (int r = 0; r < 4; ++r) {
      const float* ap =
          A + (size_t)((tr0 + r) * 16 + laneM) * D + k0 + 2 * laneH;
      af[r].x = ap[0];
      af[r].y = ap[1];
    }
    // B 4x16 fragment: rows striped across lanes; VGPR0 = K {0,2}[laneH],
    // VGPR1 = K {1,3}[laneH], col = laneM.
#pragma unroll
    for (int c = 0; c < 4; ++c) {
      const float* bp =
          B + (size_t)(k0 + 2 * laneH) * D + (tc0 + c) * 16 + laneM;
      bf[c].x = bp[0];
      bf[c].y = bp[D];
    }
#pragma unroll
    for (int r = 0; r < 4; ++r)
#pragma unroll
      for (int c = 0; c < 4; ++c)
        acc[r][c] = __builtin_amdgcn_wmma_f32_16x16x4_f32(
            false, af[r], false, bf[c], (short)0, acc[r][c], false, false);
  }

  // C/D 16x16 layout: VGPR e -> row e (lanes 0-15) / row e+8 (lanes 16-31).
#pragma unroll
  for (int r = 0; r < 4; ++r)
#pragma unroll
    for (int c = 0; c < 4; ++c)
#pragma unroll
      for (int e = 0; e < 8; ++e) {
        const int row = (tr0 + r) * 16 + e + 8 * laneH;
        const int col = (tc0 + c) * 16 + laneM;
        C[(size_t)row * D + col] = acc[r][c][e];
      }
}

// ---------------------------------------------------------------------------
// Phase 2: serial scan of the carry vector across the 64 chunk products.
//   u_0 = x ; u_{j+1} = u_j @ P_j.  Chunk products (16 MB) are L2-resident.
//   Also writes the final row res[N] = u_64.
// ---------------------------------------------------------------------------
__global__ __launch_bounds__(256) void prodscan_carries(
    const float* __restrict__ x, const float* __restrict__ ws,
    float* __restrict__ carries, float* __restrict__ out) {
  __shared__ float u[D];
  const int t = threadIdx.x;
  float v = x[t];
  u[t] = v;
  carries[t] = v;
  __syncthreads();
  for (int j = 0; j < NCHUNK; ++j) {
    const float* P = ws + (size_t)j * CHUNKSTRIDE;  // buf0 = chunk product
    if (j + 1 < NCHUNK)  // pull next chunk product ahead of the serial dep
      __builtin_prefetch(ws + (size_t)(j + 1) * CHUNKSTRIDE + t * 64, 0, 1);
    float acc = 0.f;
#pragma unroll 8
    for (int i = 0; i < D; ++i)
      acc = fmaf(u[i], P[(size_t)i * D + t], acc);  // coalesced across lanes
    __syncthreads();
    u[t] = acc;
    carries[(size_t)(j + 1) * D + t] = acc;
    __syncthreads();
  }
  out[(size_t)NMAT * D + t] = u[t];  // res[N]
}

// ---------------------------------------------------------------------------
// Phase 3: per-chunk expansion. WG j: v = u_j; emit res[16j]; then 15 GEMV
//   steps through m_{16j}..m_{16j+14}, emitting res[16j+1..16j+15].
// ---------------------------------------------------------------------------
__global__ __launch_bounds__(256) void prodscan_expand(
    const float* __restrict__ m, const float* __restrict__ carries,
    float* __restrict__ out) {
  __shared__ float vb[2][D];
  const int j = blockIdx.x;
  const int t = threadIdx.x;
  float v = carries[(size_t)j * D + t];
  vb[0][t] = v;
  out[(size_t)(j * G) * D + t] = v;
  __syncthreads();
  int cur = 0;
  for (int i = 0; i < G - 1; ++i) {
    const float* M = m + (size_t)(j * G + i) * MATELT;
    if (i + 1 < G - 1)
      __builtin_prefetch(m + (size_t)(j * G + i + 1) * MATELT + t * 64, 0, 1);
    float acc = 0.f;
#pragma unroll 8
    for (int r = 0; r < D; ++r)
      acc = fmaf(vb[cur][r], M[(size_t)r * D + t], acc);
    vb[cur ^ 1][t] = acc;                         // other buffer: no hazard
    out[(size_t)(j * G + i + 1) * D + t] = acc;
    cur ^= 1;
    __syncthreads();
  }
}

extern "C" void kernel_launch(void* const* d_in, const int* in_sizes, int n_in,
                              void* d_out, int out_size, void* d_ws,
                              size_t ws_size, hipStream_t stream) {
  (void)in_sizes; (void)n_in; (void)out_size; (void)ws_size;
  const float* x = (const float*)d_in[0];  // (1, 256)
  const float* m = (const float*)d_in[1];  // (1024, 256, 256)
  float* out = (float*)d_out;              // (1025, 256)
  float* ws  = (float*)d_ws;               // 64 chunks * (ping+pong) + carries
  float* carries = ws + (size_t)NCHUNK * CHUNKSTRIDE;  // 65 x 256 floats

  // Phase 1: 15 stream-ordered multiply steps (launch order = global sync).
  for (int t = 1; t < G; ++t)
    prodscan_chunk_mm<<<dim3(NCHUNK, 2), 256, 0, stream>>>(m, ws, t);
  // Phase 2: serial carry scan + final row.
  prodscan_carries<<<1, 256, 0, stream>>>(x, ws, carries, out);
  // Phase 3: parallel per-chunk expansion.
  prodscan_expand<<<NCHUNK, 256, 0, stream>>>(m, carries, out);
}